// FENet_19146964205728
// MI455X (gfx1250) — compile-verified
//
#include <hip/hip_runtime.h>
#include <hip/hip_bf16.h>

typedef __attribute__((ext_vector_type(2))) float v2f;
typedef __attribute__((ext_vector_type(8))) float v8f;

#define PAD_L 38

// LDS layout (floats). Strides even (8B-aligned b64 loads) and == 42 mod 64
// so the 16 rows of an A-fragment load hit distinct bank groups.
#define STRA 362                // inputs of layers 2,4,6 (max needed 360)
#define STRB 554                // inputs of layers 1,3,5 (max needed 552)
#define SZA  (16 * STRA)
#define SZB  (16 * STRB)
#define SMEM_FLOATS (SZA + SZB + 80 + 128)   // 14864 floats = 59456 B (< 64KB)

__global__ __launch_bounds__(128, 1)
void fenet_kernel(const float* __restrict__ x,
                  const float* __restrict__ fw,
                  const float* __restrict__ pw,
                  float* __restrict__ out)
{
    // per-layer geometry (input lens 900,469,254,146,92,65,52 -> outs below)
    constexpr int NT[7] = {30, 16, 10, 6, 5, 4, 3};   // ceil(L_out/16)

    extern __shared__ float smem[];
    float* bufA = smem;               // [16][STRA]
    float* bufB = smem + SZA;         // [16][STRB]
    float* wsh  = smem + SZA + SZB;   // [80]: feat w then pass w
    float* red  = wsh + 80;           // [16][8] feature sums

    const int tid  = threadIdx.x;
    const int lane = tid & 31;
    const int wv   = tid >> 5;        // wave id 0..3
    const int row  = lane & 15;       // A-row / B-col / D-col index
    const int half = lane >> 4;       // 0 or 1
    const long b0  = (long)blockIdx.x * 16;

    // B-fragment index base: idx(s,p) = 4s + p + 2*half - 2*n  (n == row)
    const int ib = (half << 1) - (row << 1);

    float psum[8];
#pragma unroll
    for (int v = 0; v < 8; ++v) psum[v] = 0.f;

    // ---------------- layer 0: A-fragments straight from global ----------------
    if (tid < 128) red[tid] = 0.f;
    if (tid < 80) wsh[tid] = (tid < 40) ? fw[tid] : pw[tid - 40];
    for (int i = tid; i < SZB; i += 128) bufB[i] = 0.f;   // layer-0 output buffer
    __syncthreads();

    {
        v2f Bf[18], Bp[18];
#pragma unroll
        for (int s = 0; s < 18; ++s) {
            int i0 = 4 * s + ib, i1 = i0 + 1;
            int c0 = min(max(i0, 0), 39), c1 = min(max(i1, 0), 39);
            float f0 = wsh[c0],      f1 = wsh[c1];
            float p0 = wsh[40 + c0], p1 = wsh[40 + c1];
            bool ok0 = (i0 >= 0) && (i0 < 40);
            bool ok1 = (i1 >= 0) && (i1 < 40);
            Bf[s].x = ok0 ? f0 : 0.f;  Bf[s].y = ok1 ? f1 : 0.f;
            Bp[s].x = ok0 ? p0 : 0.f;  Bp[s].y = ok1 ? p1 : 0.f;
        }

        float fsum[8];
#pragma unroll
        for (int v = 0; v < 8; ++v) fsum[v] = 0.f;

        const float* xr = x + (b0 + row) * 900;   // this lane's input row

        for (int t = wv; t < 30; t += 4) {
            const int t0 = t << 4;
            const int gb = (t0 << 1) + (half << 1) - PAD_L;  // data index, even

            v8f accf = {0.f,0.f,0.f,0.f,0.f,0.f,0.f,0.f};
            v8f accp = {0.f,0.f,0.f,0.f,0.f,0.f,0.f,0.f};

            if (t >= 2 && t <= 27) {
                // interior: window [gb, gb+69] is fully inside [0,900)
                const float* ap = xr + gb;
#pragma unroll
                for (int s = 0; s < 18; ++s) {
                    v2f a = *(const v2f*)(ap + 4 * s);  // global_load_b64 +imm
                    accf = __builtin_amdgcn_wmma_f32_16x16x4_f32(
                        false, a, false, Bf[s], (short)0, accf, false, false);
                    accp = __builtin_amdgcn_wmma_f32_16x16x4_f32(
                        false, a, false, Bp[s], (short)0, accp, false, false);
                }
            } else {
                // edge tiles (pads): clamped address + per-component select
#pragma unroll
                for (int s = 0; s < 18; ++s) {
                    int g0 = gb + 4 * s;
                    int gc = min(max(g0, 0), 898);           // stays even
                    v2f d  = *(const v2f*)(xr + gc);         // aligned b64
                    v2f a;
                    a.x = ((unsigned)g0       < 900u) ? d.x : 0.f;
                    a.y = ((unsigned)(g0 + 1) < 900u) ? d.y : 0.f;
                    accf = __builtin_amdgcn_wmma_f32_16x16x4_f32(
                        false, a, false, Bf[s], (short)0, accf, false, false);
                    accp = __builtin_amdgcn_wmma_f32_16x16x4_f32(
                        false, a, false, Bp[s], (short)0, accp, false, false);
                }
            }

#pragma unroll
            for (int v = 0; v < 8; ++v) fsum[v] += __builtin_fabsf(accf[v]);
            const int colbase = PAD_L + t0 + row;
#pragma unroll
            for (int v = 0; v < 8; ++v) {
                int m = v + (half << 3);
                bufB[m * STRB + colbase] = accp[v];
            }
        }
#pragma unroll
        for (int v = 0; v < 8; ++v) {
            int m = v + (half << 3);
            atomicAdd(&red[m * 8 + 0], fsum[v]);
        }
    }

    // ---------------- layers 1..6: LDS ping-pong ----------------
    for (int ell = 1; ell < 7; ++ell) {
        const int nt    = NT[ell];
        const int sIn   = (ell & 1) ? STRB : STRA;
        float* inb      = (ell & 1) ? bufB : bufA;
        float* outb     = (ell & 1) ? bufA : bufB;
        const int sOut  = (ell & 1) ? STRA : STRB;
        const int szOut = (ell & 1) ? SZA  : SZB;

        __syncthreads();   // previous layer fully consumed/produced
        if (tid < 80) wsh[tid] = (tid < 40) ? fw[ell * 40 + tid]
                                            : pw[ell * 40 + tid - 40];
        if (ell < 6)
            for (int i = tid; i < szOut; i += 128) outb[i] = 0.f;
        __syncthreads();

        v2f Bf[18], Bp[18];
#pragma unroll
        for (int s = 0; s < 18; ++s) {
            int i0 = 4 * s + ib, i1 = i0 + 1;
            int c0 = min(max(i0, 0), 39), c1 = min(max(i1, 0), 39);
            float f0 = wsh[c0],      f1 = wsh[c1];
            float p0 = wsh[40 + c0], p1 = wsh[40 + c1];
            bool ok0 = (i0 >= 0) && (i0 < 40);
            bool ok1 = (i1 >= 0) && (i1 < 40);
            Bf[s].x = ok0 ? f0 : 0.f;  Bf[s].y = ok1 ? f1 : 0.f;
            Bp[s].x = ok0 ? p0 : 0.f;  Bp[s].y = ok1 ? p1 : 0.f;
        }

        float fsum[8];
#pragma unroll
        for (int v = 0; v < 8; ++v) fsum[v] = 0.f;

        for (int t = wv; t < nt; t += 4) {
            const int t0 = t << 4;
            const float* ap = inb + row * sIn + (t0 << 1) + (half << 1);

            v8f accf = {0.f,0.f,0.f,0.f,0.f,0.f,0.f,0.f};
            v8f accp = {0.f,0.f,0.f,0.f,0.f,0.f,0.f,0.f};
#pragma unroll
            for (int s = 0; s < 18; ++s) {
                v2f a = *(const v2f*)(ap + 4 * s);   // ds_load_b64 (fused 2addr)
                accf = __builtin_amdgcn_wmma_f32_16x16x4_f32(
                    false, a, false, Bf[s], (short)0, accf, false, false);
                accp = __builtin_amdgcn_wmma_f32_16x16x4_f32(
                    false, a, false, Bp[s], (short)0, accp, false, false);
            }

            // feat branch: LeakyReLU(-1) == |x|; t >= L_out columns are exact 0
#pragma unroll
            for (int v = 0; v < 8; ++v) fsum[v] += __builtin_fabsf(accf[v]);

            if (ell < 6) {
                const int colbase = PAD_L + t0 + row;
#pragma unroll
                for (int v = 0; v < 8; ++v) {
                    int m = v + (half << 3);
                    outb[m * sOut + colbase] = accp[v];
                }
            } else {
#pragma unroll
                for (int v = 0; v < 8; ++v) psum[v] += __builtin_fabsf(accp[v]);
            }
        }

#pragma unroll
        for (int v = 0; v < 8; ++v) {
            int m = v + (half << 3);
            atomicAdd(&red[m * 8 + ell], fsum[v]);
        }
    }

    // final bitshift-pool feature from last pass output
#pragma unroll
    for (int v = 0; v < 8; ++v) {
        int m = v + (half << 3);
        atomicAdd(&red[m * 8 + 7], psum[v]);
    }
    __syncthreads();

    if (tid < 128) {
        constexpr float inv[8] = {1.f/469.f, 1.f/254.f, 1.f/146.f, 1.f/92.f,
                                  1.f/65.f,  1.f/52.f,  1.f/45.f,  1.f/32.f};
        int m = tid >> 3, i = tid & 7;
        out[(b0 + m) * 8 + i] = red[tid] * inv[i];
    }
}

extern "C" void kernel_launch(void* const* d_in, const int* in_sizes, int n_in,
                              void* d_out, int out_size, void* d_ws, size_t ws_size,
                              hipStream_t stream)
{
    const float* x  = (const float*)d_in[0];
    const float* fw = (const float*)d_in[1];
    const float* pw = (const float*)d_in[2];
    float* outp     = (float*)d_out;

    const int batch  = in_sizes[0] / 900;          // 24576
    const int blocks = batch / 16;                 // 1536 workgroups of 4 waves
    const size_t smem = SMEM_FLOATS * sizeof(float);

    hipLaunchKernelGGL(fenet_kernel, dim3(blocks), dim3(128), smem, stream,
                       x, fw, pw, outp);
}